// Anchors_30210799960227
// MI455X (gfx1250) — compile-verified
//
#include <hip/hip_runtime.h>

namespace {
constexpr int   kNGT = 64;
constexpr float kFG  = 0.5f;
constexpr float kBG  = 0.4f;
}

// ---- CDNA5 async global->LDS path (signatures verified by compile probes) ----
#if __has_builtin(__builtin_amdgcn_global_load_async_to_lds_b128) && \
    __has_builtin(__builtin_amdgcn_global_load_async_to_lds_b32)  && \
    __has_builtin(__builtin_amdgcn_s_wait_asynccnt)
#define USE_ASYNC_LDS 1
#else
#define USE_ASYNC_LDS 0
#endif

#if USE_ASYNC_LDS
typedef int v4i __attribute__((vector_size(16)));   // matches builtin's int4 pointee
#define AS1 __attribute__((address_space(1)))
#define AS3 __attribute__((address_space(3)))

__device__ __forceinline__ AS1 v4i* as_global_b128(const void* p) {
  return (AS1 v4i*)(unsigned long long)p;
}
__device__ __forceinline__ AS1 int* as_global_b32(const void* p) {
  return (AS1 int*)(unsigned long long)p;
}
__device__ __forceinline__ AS3 v4i* as_lds_b128(const void* p) {
  // generic LDS pointer: low 32 bits are the LDS byte offset (aperture is in [63:32])
  return (AS3 v4i*)(unsigned int)(unsigned long long)p;
}
__device__ __forceinline__ AS3 int* as_lds_b32(const void* p) {
  return (AS3 int*)(unsigned int)(unsigned long long)p;
}
#endif

__global__ __launch_bounds__(256) void anchor_assign_kernel(
    const float4* __restrict__ anchors,       // (NA,4)  cx cy w h
    const float4* __restrict__ anchors_xyxy,  // (NA,4)  x1 y1 x2 y2
    const float4* __restrict__ gt_boxes,      // (B,64,4) xyxy
    const int*    __restrict__ gt_labels,     // (B,64)
    float*        __restrict__ loc_out,       // (B,NA,4)
    float*        __restrict__ cls_out,       // (B,NA)
    int na)
{
  __shared__ float4 s_box[kNGT];
  __shared__ int    s_lab[kNGT];
  __shared__ float2 s_meta[kNGT];  // x = gt area, y = valid flag (1/0)

  const int tid  = threadIdx.x;
  const int b    = (int)blockIdx.y;
  const int base = (int)blockIdx.x * 512;
  const int a0   = base + tid;        // first anchor for this lane
  const int a1   = a0 + 256;          // second anchor for this lane
  const bool v0  = a0 < na;
  const bool v1  = a1 < na;

  // Early-issue this lane's anchor loads so their latency hides behind the
  // GT-box async DMA + barriers below.
  float4 axy0  = make_float4(0.f, 0.f, 1.f, 1.f);
  float4 actr0 = make_float4(0.f, 0.f, 1.f, 1.f);
  if (v0) { axy0 = anchors_xyxy[a0]; actr0 = anchors[a0]; }
  float4 axy1  = axy0;
  float4 actr1 = actr0;
  if (v1) { axy1 = anchors_xyxy[a1]; actr1 = anchors[a1]; }

#if USE_ASYNC_LDS
  if (tid < kNGT) {
    __builtin_amdgcn_global_load_async_to_lds_b128(
        as_global_b128(&gt_boxes[(size_t)b * kNGT + tid]),
        as_lds_b128(&s_box[tid]), 0, 0);
    __builtin_amdgcn_global_load_async_to_lds_b32(
        as_global_b32(&gt_labels[(size_t)b * kNGT + tid]),
        as_lds_b32(&s_lab[tid]), 0, 0);
  }
  __builtin_amdgcn_s_wait_asynccnt(0);
  __syncthreads();
#else
  if (tid < kNGT) {
    s_box[tid] = gt_boxes[(size_t)b * kNGT + tid];
    s_lab[tid] = gt_labels[(size_t)b * kNGT + tid];
  }
  __syncthreads();
#endif

  if (tid < kNGT) {
    const float4 g = s_box[tid];
    s_meta[tid] = make_float2((g.z - g.x) * (g.w - g.y),
                              (s_lab[tid] != -2) ? 1.0f : 0.0f);
  }
  __syncthreads();

  if (!v0) return;

  const float area_a0 = (axy0.z - axy0.x) * (axy0.w - axy0.y);
  const float area_a1 = (axy1.z - axy1.x) * (axy1.w - axy1.y);

  // Division-free argmax over iou = inter/denom (denom > 0).
  // Padded GT behaves as score -1/1 (matches reference where(pad, -1, iou));
  // strict-greater scan reproduces argmax's first-max tie rule.
  float bn0 = -2.0f, bd0 = 1.0f;  int bi0 = 0;
  float bn1 = -2.0f, bd1 = 1.0f;  int bi1 = 0;
#pragma unroll 8
  for (int m = 0; m < kNGT; ++m) {
    const float4 g  = s_box[m];     // one ds_load_b128 feeds both anchors
    const float2 md = s_meta[m];    // one ds_load_b64 feeds both anchors
    const bool valid = (md.y != 0.0f);

    // anchor 0
    {
      float w = fminf(axy0.z, g.z) - fmaxf(axy0.x, g.x);
      float h = fminf(axy0.w, g.w) - fmaxf(axy0.y, g.y);
      w = fmaxf(w, 0.0f);
      h = fmaxf(h, 0.0f);
      float inter = w * h;
      float denom = (area_a0 + md.x) - inter;
      inter = valid ? inter : -1.0f;
      denom = valid ? denom : 1.0f;
      const bool better = inter * bd0 > bn0 * denom;
      bn0 = better ? inter : bn0;
      bd0 = better ? denom : bd0;
      bi0 = better ? m     : bi0;
    }
    // anchor 1
    {
      float w = fminf(axy1.z, g.z) - fmaxf(axy1.x, g.x);
      float h = fminf(axy1.w, g.w) - fmaxf(axy1.y, g.y);
      w = fmaxf(w, 0.0f);
      h = fmaxf(h, 0.0f);
      float inter = w * h;
      float denom = (area_a1 + md.x) - inter;
      inter = valid ? inter : -1.0f;
      denom = valid ? denom : 1.0f;
      const bool better = inter * bd1 > bn1 * denom;
      bn1 = better ? inter : bn1;
      bd1 = better ? denom : bd1;
      bi1 = better ? m     : bi1;
    }
  }

  auto finish = [&](float bn, float bd, int bi, const float4& actr, size_t o) {
    const float best_iou = bn / bd;  // single divide per anchor
    int label = s_lab[bi];
    if (best_iou > kBG && best_iou < kFG) label = -1;  // ignore band
    if (best_iou < kBG)                   label = 0;   // background

    const float4 g  = s_box[bi];
    const float bcx = (g.x + g.z) * 0.5f;
    const float bcy = (g.y + g.w) * 0.5f;
    const float bw  = g.z - g.x;
    const float bh  = g.w - g.y;
    const float dx  = (bcx - actr.x) / (actr.z * 0.1f);
    const float dy  = (bcy - actr.y) / (actr.w * 0.1f);
    const float dw  = __logf(bw / actr.z) * 5.0f;   // / 0.2
    const float dh  = __logf(bh / actr.w) * 5.0f;

    reinterpret_cast<float4*>(loc_out)[o] = make_float4(dx, dy, dw, dh);
    cls_out[o] = (float)label;
  };

  const size_t o0 = (size_t)b * (size_t)na + (size_t)a0;
  finish(bn0, bd0, bi0, actr0, o0);
  if (v1) finish(bn1, bd1, bi1, actr1, o0 + 256);
}

extern "C" void kernel_launch(void* const* d_in, const int* in_sizes, int n_in,
                              void* d_out, int out_size, void* d_ws, size_t ws_size,
                              hipStream_t stream) {
  (void)n_in; (void)out_size; (void)d_ws; (void)ws_size;
  const float4* anchors      = (const float4*)d_in[0];
  const float4* anchors_xyxy = (const float4*)d_in[1];
  const float4* gt_boxes     = (const float4*)d_in[2];
  const int*    gt_labels    = (const int*)d_in[3];

  const int na = in_sizes[0] / 4;          // 76500 anchors
  const int nb = in_sizes[3] / kNGT;       // 16 batches

  float* loc_out = (float*)d_out;                          // (B,NA,4)
  float* cls_out = loc_out + (size_t)nb * (size_t)na * 4;  // (B,NA)

  dim3 block(256, 1, 1);
  dim3 grid((unsigned)((na + 511) / 512), (unsigned)nb, 1);
  anchor_assign_kernel<<<grid, block, 0, stream>>>(
      anchors, anchors_xyxy, gt_boxes, gt_labels, loc_out, cls_out, na);
}